// MIPSBruteTopK_74594991997474
// MI455X (gfx1250) — compile-verified
//
#include <hip/hip_runtime.h>

typedef float v2f __attribute__((ext_vector_type(2)));
typedef float v8f __attribute__((ext_vector_type(8)));

#define B_Q        256
#define D_DIM      128
#define X_ITEMS    1000000
#define K_TOP      100
#define BT         64                       // items per big tile (staged in LDS)
#define NBT_TOTAL  (X_ITEMS / BT)           // 15625 (exact)
#define CHUNK_BT   32                       // big tiles per chunk -> 2048 items
#define NUM_CHUNKS ((NBT_TOTAL + CHUNK_BT - 1) / CHUNK_BT)  // 489
#define CPC        8                        // candidates kept per (query, chunk)
#define CAND_PER_Q (NUM_CHUNKS * CPC)       // 3912
#define NPAD       4096
#define LDSB_ST    68                       // padded row stride (floats) for item tile
#define LDSB_SZ    (D_DIM * LDSB_ST)        // floats per buffer
#define CT_ST      18                       // padded row stride (floats) for C tile

#if __has_builtin(__builtin_amdgcn_s_wait_asynccnt)
#define ASYNC_WAIT(n) __builtin_amdgcn_s_wait_asynccnt(n)
#else
#define ASYNC_WAIT(n) asm volatile("s_wait_asynccnt %0" :: "n"(n) : "memory")
#endif

__device__ __forceinline__ void ins8(float (&ts)[8], int (&ti)[8], float cs, int ci) {
    #pragma unroll
    for (int p = 0; p < 8; ++p) {
        bool sw = cs > ts[p];
        float os = ts[p]; int oi = ti[p];
        ts[p] = sw ? cs : os;  ti[p] = sw ? ci : oi;
        cs    = sw ? os : cs;  ci    = sw ? oi : ci;
    }
}

__global__ __launch_bounds__(512) void k_gemm_topc(
    const float* __restrict__ Q, const float* __restrict__ It,
    float* __restrict__ candS, int* __restrict__ candI)
{
    __shared__ float ldsB[2 * LDSB_SZ];            // double-buffered item tile, 69,632 B
    __shared__ float ctile[16 * 16 * CT_ST];       // per-wave 16x16 slabs, 18,432 B

    const int tid  = threadIdx.x;
    const int lane = tid & 31;
    const int wave = tid >> 5;                     // query block 0..15
    const int chunk = blockIdx.x;
    const int bt0  = chunk * CHUNK_BT;
    const int nbt  = min(CHUNK_BT, NBT_TOTAL - bt0);

    const int m  = lane & 15;                      // row within 16 (A) / col within 16 (B)
    const int kh = (lane >> 4) * 2;                // K sub-offset per lane half

    // ---- A tile (16 queries x 128 dims) resident in 64 VGPRs ----
    v2f a[32];
    const float* qrow = Q + (size_t)(wave * 16 + m) * D_DIM + kh;
    #pragma unroll
    for (int k4 = 0; k4 < 32; ++k4)
        a[k4] = *(const v2f*)(qrow + 4 * k4);

    // ---- per-lane running top-8 for one query (half the columns) ----
    float ts[8]; int ti[8];
    #pragma unroll
    for (int j = 0; j < 8; ++j) { ts[j] = -INFINITY; ti[j] = 0x7fffffff; }

    float* myct = ctile + wave * (16 * CT_ST);

    // ---- async staging: each lane copies 4x16B of the 64x128 tile, no VGPR payload ----
    const unsigned ldsBase = (unsigned)(unsigned long long)&ldsB[0];
    const int sidx = tid;                          // 512 threads cover 2048 float4s in 4 steps
    auto issueTile = [&](int bt, int buf) {
        const int col0 = bt * BT;
        const unsigned base = ldsBase + (unsigned)buf * (LDSB_SZ * 4u);
        #pragma unroll
        for (int i = 0; i < 4; ++i) {
            int idx = i * 512 + sidx;
            int row = idx >> 4, c4 = idx & 15;
            unsigned dst = base + (unsigned)((row * LDSB_ST + c4 * 4) * 4);
            const float* src = It + (size_t)row * X_ITEMS + col0 + c4 * 4;
            asm volatile("global_load_async_to_lds_b128 %0, %1, off"
                         :: "v"(dst), "v"(src) : "memory");
        }
    };

    issueTile(bt0, 0);
    for (int t = 0; t < nbt; ++t) {
        if (t + 1 < nbt) {                         // overlap next tile's DMA with WMMA
            issueTile(bt0 + t + 1, (t + 1) & 1);
            ASYNC_WAIT(4);                         // in-order: tile t complete, t+1 in flight
        } else {
            ASYNC_WAIT(0);
        }
        __syncthreads();

        const float* Bb = ldsB + (t & 1) * LDSB_SZ;
        const int itemBase = (bt0 + t) * BT;
        #pragma unroll
        for (int s = 0; s < 4; ++s) {              // four 16-column sub-tiles
            v8f acc;
            #pragma unroll
            for (int z = 0; z < 8; ++z) acc[z] = 0.0f;

            const int n = s * 16 + m;              // B column for this lane
            // B pipeline, 2 K-steps ahead
            v2f b0, b1;
            b0.x = Bb[(kh    ) * LDSB_ST + n];  b0.y = Bb[(kh + 1) * LDSB_ST + n];
            b1.x = Bb[(kh + 4) * LDSB_ST + n];  b1.y = Bb[(kh + 5) * LDSB_ST + n];
            #pragma unroll
            for (int k4 = 0; k4 < 32; ++k4) {
                v2f bn = b1;
                if (k4 < 30) {
                    const int k0 = 4 * (k4 + 2) + kh;
                    bn.x = Bb[k0 * LDSB_ST + n];
                    bn.y = Bb[(k0 + 1) * LDSB_ST + n];
                }
                acc = __builtin_amdgcn_wmma_f32_16x16x4_f32(
                        false, a[k4], false, b0, (short)0, acc, false, false);
                b0 = b1; b1 = bn;
            }

            // transpose C through per-wave LDS slab
            const int mrow = (lane >> 4) * 8;
            #pragma unroll
            for (int r = 0; r < 8; ++r)
                myct[(mrow + r) * CT_ST + m] = acc[r];

            // lane scans its query row (m) over its half of columns
            const int nb = (lane >> 4) * 8;
            #pragma unroll
            for (int jj = 0; jj < 8; ++jj) {
                float sc = myct[m * CT_ST + nb + jj];
                if (sc > ts[7])
                    ins8(ts, ti, sc, itemBase + s * 16 + nb + jj);
            }
        }
        __syncthreads();                            // protects buf[(t+1)&1] reuse next iter
    }

    // merge lane pairs (l, l^16) -> exact top-8 per query over the chunk
    float ps[8]; int pi[8];
    #pragma unroll
    for (int j = 0; j < 8; ++j) {
        ps[j] = __shfl(ts[j], lane ^ 16, 32);
        pi[j] = __shfl(ti[j], lane ^ 16, 32);
    }
    #pragma unroll
    for (int j = 0; j < 8; ++j)
        if (ps[j] > ts[7]) ins8(ts, ti, ps[j], pi[j]);

    if (lane < 16) {
        const int q = wave * 16 + lane;
        const size_t base = ((size_t)q * NUM_CHUNKS + chunk) * CPC;
        #pragma unroll
        for (int j = 0; j < CPC; ++j) { candS[base + j] = ts[j]; candI[base + j] = ti[j]; }
    }
}

__global__ __launch_bounds__(256) void k_select(
    const float* __restrict__ candS, const int* __restrict__ candI,
    const long long* __restrict__ ids,
    float* __restrict__ outL, long long* __restrict__ outI)
{
    __shared__ float ssc[NPAD];
    __shared__ int   six[NPAD];
    __shared__ unsigned long long rkey[256];
    __shared__ int   rpos[256];

    const int q = blockIdx.x, tid = threadIdx.x;
    for (int i = tid; i < NPAD; i += 256) {
        if (i < CAND_PER_Q) {
            ssc[i] = candS[(size_t)q * CAND_PER_Q + i];
            six[i] = candI[(size_t)q * CAND_PER_Q + i];
        } else { ssc[i] = -INFINITY; six[i] = 0x7fffffff; }
    }
    __syncthreads();

    for (int r = 0; r < K_TOP; ++r) {
        unsigned long long bk = 0; int bp = 0;
        #pragma unroll 4
        for (int i = tid * 16; i < tid * 16 + 16; ++i) {
            unsigned u = __float_as_uint(ssc[i]);
            u ^= (u & 0x80000000u) ? 0xffffffffu : 0x80000000u;   // monotone key
            unsigned long long key =
                ((unsigned long long)u << 32) | (unsigned)(~six[i]); // tie: smaller idx wins
            if (key > bk) { bk = key; bp = i; }
        }
        rkey[tid] = bk; rpos[tid] = bp;
        __syncthreads();
        for (int off = 128; off > 0; off >>= 1) {
            if (tid < off && rkey[tid + off] > rkey[tid]) {
                rkey[tid] = rkey[tid + off]; rpos[tid] = rpos[tid + off];
            }
            __syncthreads();
        }
        if (tid == 0) {
            int p = rpos[0];
            outL[q * K_TOP + r] = ssc[p];
            outI[q * K_TOP + r] = ids[six[p]];
            ssc[p] = -INFINITY; six[p] = 0x7fffffff;
        }
        __syncthreads();
    }
}

extern "C" void kernel_launch(void* const* d_in, const int* in_sizes, int n_in,
                              void* d_out, int out_size, void* d_ws, size_t ws_size,
                              hipStream_t stream) {
    const float*     Qe  = (const float*)d_in[0];       // [256,128]
    const float*     It  = (const float*)d_in[1];       // [128,1e6]
    const long long* ids = (const long long*)d_in[2];   // [1,1e6] int64
    // d_in[3] = k (==100), baked in at compile time

    float* candS = (float*)d_ws;
    int*   candI = (int*)((char*)d_ws + (size_t)B_Q * CAND_PER_Q * sizeof(float));

    float*     outL = (float*)d_out;
    long long* outI = (long long*)((char*)d_out + (size_t)B_Q * K_TOP * sizeof(float));

    hipLaunchKernelGGL(k_gemm_topc, dim3(NUM_CHUNKS), dim3(512), 0, stream,
                       Qe, It, candS, candI);
    hipLaunchKernelGGL(k_select, dim3(B_Q), dim3(256), 0, stream,
                       candS, candI, ids, outL, outI);
}